// UnimodalConcentratedLoss_64939905516212
// MI455X (gfx1250) — compile-verified
//
#include <hip/hip_runtime.h>
#include <hip/hip_bf16.h>

// ---------------------------------------------------------------------------
// UnimodalConcentratedLoss  (B=262144 rows, C=101 classes, fp32)
//   out[0] = total, out[1] = concentrated, out[2] = LAMBDA * unimodal
//
// Bandwidth-bound: one coalesced pass over 106 MB of logits (~4.5us at
// 23.3 TB/s).  Wave-per-row softmax stats with wave32 shuffles; deterministic
// two-pass reduction (per-block partials -> single-wave WMMA reduction).
// ---------------------------------------------------------------------------

#define NUM_CLASSES 101
#define BATCH       262144
#define LAMBDA_W    1000.0f

#define ROWS_PER_BLOCK 64   // 8 waves * 8 rows
#define NUM_BLOCKS     (BATCH / ROWS_PER_BLOCK)   // 4096

typedef __attribute__((ext_vector_type(2))) float v2f;
typedef __attribute__((ext_vector_type(8))) float v8f;

__device__ __forceinline__ float wave_sum(float v) {
    #pragma unroll
    for (int o = 16; o; o >>= 1) v += __shfl_xor(v, o, 32);
    return v;
}
__device__ __forceinline__ float wave_max(float v) {
    #pragma unroll
    for (int o = 16; o; o >>= 1) v = fmaxf(v, __shfl_xor(v, o, 32));
    return v;
}

__global__ __launch_bounds__(256)
void ucl_main_kernel(const float* __restrict__ logits,
                     const int*   __restrict__ targets,
                     float*       __restrict__ partials)   // [2 * NUM_BLOCKS]
{
    const int lane = threadIdx.x & 31;
    const int wave = threadIdx.x >> 5;

    __shared__ float sC[8];
    __shared__ float sP[8];

    float concAcc = 0.0f;
    float penAcc  = 0.0f;

    const int rowBase = blockIdx.x * ROWS_PER_BLOCK;

    #pragma unroll 1
    for (int r = 0; r < 8; ++r) {
        const int row = rowBase + r * 8 + wave;   // 8 waves cover 8 consecutive rows
        const float* x = logits + (size_t)row * NUM_CLASSES;

        // prefetch next row (emits global_prefetch_b8)
        __builtin_prefetch(x + NUM_CLASSES, 0, 1);

        // ---- load 4 logits per lane (classes lane, +32, +64, +96) ----------
        const float v0 = x[lane];
        const float v1 = x[lane + 32];
        const float v2 = x[lane + 64];
        const float v3 = (lane < NUM_CLASSES - 96) ? x[lane + 96] : -3.0e38f;

        // ---- softmax numerator --------------------------------------------
        const float m = wave_max(fmaxf(fmaxf(v0, v1), fmaxf(v2, v3)));

        float e[5];
        e[0] = __expf(v0 - m);
        e[1] = __expf(v1 - m);
        e[2] = __expf(v2 - m);
        e[3] = (lane < NUM_CLASSES - 96) ? __expf(v3 - m) : 0.0f;
        e[4] = 0.0f;

        const float c0 = (float)lane;
        const float c1 = (float)(lane + 32);
        const float c2 = (float)(lane + 64);
        const float c3 = (float)(lane + 96);

        float Z  = e[0] + e[1] + e[2] + e[3];
        float S1 = e[0] * c0 + e[1] * c1 + e[2] * c2 + e[3] * c3;
        float S2 = e[0] * c0 * c0 + e[1] * c1 * c1 + e[2] * c2 * c2 + e[3] * c3 * c3;

        Z  = wave_sum(Z);
        S1 = wave_sum(S1);
        S2 = wave_sum(S2);

        const float invZ = 1.0f / Z;
        const float pred = S1 * invZ;
        float       var  = fmaxf(S2 * invZ - pred * pred, 1e-6f);

        const int   t  = targets[row];
        const float d  = pred - (float)t;
        const float conc = 0.5f * __logf(var) + d * d / (2.0f * var);

        // ---- unimodal penalty: sum_j relu(-(p_j - p_{j+1}) * sign_j) -------
        float pen = 0.0f;
        #pragma unroll
        for (int s = 0; s < 4; ++s) {
            const int j = lane + 32 * s;
            // e_{j+1}: lane+1 same slot, or lane0 of next slot when lane==31
            const float nA = __shfl(e[s],     (lane + 1) & 31, 32);
            const float nB = __shfl(e[s + 1], 0,              32);
            const float enext = (lane == 31) ? nB : nA;
            const float diff  = e[s] - enext;
            const float sgn   = (j < t) ? -1.0f : 1.0f;
            const float term  = fmaxf(0.0f, -diff * sgn);
            if (j < NUM_CLASSES - 1) pen += term;
        }
        pen = wave_sum(pen) * invZ;

        concAcc += conc;
        penAcc  += pen;
    }

    if (lane == 0) { sC[wave] = concAcc; sP[wave] = penAcc; }
    __syncthreads();

    if (threadIdx.x == 0) {
        float cc = 0.0f, pp = 0.0f;
        #pragma unroll
        for (int w = 0; w < 8; ++w) { cc += sC[w]; pp += sP[w]; }
        partials[blockIdx.x]              = cc;   // fixed slot: deterministic
        partials[NUM_BLOCKS + blockIdx.x] = pp;
    }
}

// ---------------------------------------------------------------------------
// Final reduction: single wave32.  Sums 2 x NUM_BLOCKS partials through
// V_WMMA_F32_16X16X4_F32 with an all-ones B matrix.  With B = ones, the sum
// of ALL 256 entries of D equals 16 * sum(A) regardless of the exact lane
// packing of A, so this is layout-proof and bitwise deterministic.
// ---------------------------------------------------------------------------
__global__ __launch_bounds__(32)
void ucl_reduce_kernel(const float* __restrict__ partials,
                       float*       __restrict__ out)
{
    const int lane = threadIdx.x;   // 0..31, full wave => EXEC all ones

    const float* conc = partials;
    const float* pen  = partials + NUM_BLOCKS;

    v8f accC = {};
    v8f accP = {};
    v2f ones; ones.x = 1.0f; ones.y = 1.0f;

    const int chunks = NUM_BLOCKS / 64;   // 64 values of A per WMMA step

    #pragma unroll 1
    for (int i = 0; i < chunks; ++i) {
        const int base = i * 64;
        v2f a, b;
        a.x = conc[base + lane];
        a.y = conc[base + 32 + lane];
        b.x = pen [base + lane];
        b.y = pen [base + 32 + lane];

        accC = __builtin_amdgcn_wmma_f32_16x16x4_f32(
            false, a, false, ones, (short)0, accC, false, false);
        accP = __builtin_amdgcn_wmma_f32_16x16x4_f32(
            false, b, false, ones, (short)0, accP, false, false);
    }

    float c = 0.0f, p = 0.0f;
    #pragma unroll
    for (int g = 0; g < 8; ++g) { c += accC[g]; p += accP[g]; }
    c = wave_sum(c) * (1.0f / 16.0f);   // sum(D) == 16 * sum(partials)
    p = wave_sum(p) * (1.0f / 16.0f);

    if (lane == 0) {
        const float conc_loss = c / (float)BATCH;
        const float uni_loss  = LAMBDA_W * (p / (float)BATCH);
        out[0] = conc_loss + uni_loss;
        out[1] = conc_loss;
        out[2] = uni_loss;
    }
}

extern "C" void kernel_launch(void* const* d_in, const int* in_sizes, int n_in,
                              void* d_out, int out_size, void* d_ws, size_t ws_size,
                              hipStream_t stream) {
    const float* logits  = (const float*)d_in[0];
    const int*   targets = (const int*)d_in[1];
    float*       wsf     = (float*)d_ws;          // needs 2*NUM_BLOCKS floats = 32 KB
    float*       outf    = (float*)d_out;

    ucl_main_kernel<<<NUM_BLOCKS, 256, 0, stream>>>(logits, targets, wsf);
    ucl_reduce_kernel<<<1, 32, 0, stream>>>(wsf, outf);
}